// HeteroGNN_56092272886142
// MI455X (gfx1250) — compile-verified
//
#include <hip/hip_runtime.h>
#include <hip/hip_bf16.h>

// ---------------------------------------------------------------------------
// HeteroGNN (2-layer bipartite GraphSAGE) for MI455X / gfx1250.
// Bandwidth-bound on edge gather/scatter (~9 GB); dense transforms use
// full-precision fp32 WMMA (V_WMMA_F32_16X16X4_F32).
// ---------------------------------------------------------------------------

#define N_USER  200000
#define N_MOVIE 100000
#define N_EDGE  4000000
#define D       64

typedef float v2f __attribute__((ext_vector_type(2)));
typedef float v8f __attribute__((ext_vector_type(8)));

// LDS strides chosen for bank-conflict-free wave32 access (64 banks):
//   weight B-fragment: lanes 0-15 read row k, lanes 16-31 row k+2.
//   stride 80 -> rows 2 apart shift banks by (2*80)%64 = 32 -> disjoint 16-bank sets.
#define WSTRIDE 80
//   A-fragment: lanes 0-15 read 16 different rows at same column pair.
//   stride 68 -> consecutive rows shift banks by 4 -> all lanes distinct banks.
#define ASTRIDE 68

// ---------------------------------------------------------------------------
// Workspace layout (floats)
// ---------------------------------------------------------------------------
#define OFF_SU 0u                         // [N_USER  x 64] segment sums (user dst)
#define OFF_SM (OFF_SU + N_USER * 64u)    // [N_MOVIE x 64] segment sums (movie dst)
#define OFF_CU (OFF_SM + N_MOVIE * 64u)   // [N_USER]  degree counts
#define OFF_CM (OFF_CU + N_USER)          // [N_MOVIE] degree counts
#define OFF_U1 (OFF_CM + N_MOVIE)         // [N_USER  x 64] layer-1 user output
#define OFF_M1 (OFF_U1 + N_USER * 64u)    // [N_MOVIE x 64] layer-1 movie output
#define ZERO_FLOATS (OFF_U1)              // sums + counts region, zeroed per layer
#define ZERO_F4 (ZERO_FLOATS / 4u)        // 19,500,000 / 4

// ---------------------------------------------------------------------------
// Kernel 1: zero a region with float4 stores
// ---------------------------------------------------------------------------
__global__ __launch_bounds__(256) void zero_f4_kernel(float4* __restrict__ p,
                                                      unsigned n4) {
  unsigned i = blockIdx.x * 256u + threadIdx.x;
  if (i < n4) p[i] = make_float4(0.f, 0.f, 0.f, 0.f);
}

// ---------------------------------------------------------------------------
// Kernel 2: edge scatter for BOTH relations. One wave32 per edge; each lane
// owns 2 of the 64 features. Atomic f32 adds land in L2 (sum buffers < L2).
// ---------------------------------------------------------------------------
__global__ __launch_bounds__(256) void scatter_kernel(
    const float* __restrict__ xu, const float* __restrict__ xm,
    const int* __restrict__ uidx, const int* __restrict__ midx,
    float* __restrict__ su, float* __restrict__ sm,
    float* __restrict__ cu, float* __restrict__ cm, int nedge) {
  const int e    = (int)((blockIdx.x * 256u + threadIdx.x) >> 5);
  const int lane = threadIdx.x & 31;
  if (e >= nedge) return;
  const int u = uidx[e];
  const int m = midx[e];
  const int c = lane * 2;
  // relation user -> movie : gather x_user[u], accumulate into s_movie[m]
  const float2 a = *(const float2*)&xu[(size_t)u * D + c];
  atomicAdd(&sm[(size_t)m * D + c + 0], a.x);
  atomicAdd(&sm[(size_t)m * D + c + 1], a.y);
  // relation movie -> user : gather x_movie[m], accumulate into s_user[u]
  const float2 b = *(const float2*)&xm[(size_t)m * D + c];
  atomicAdd(&su[(size_t)u * D + c + 0], b.x);
  atomicAdd(&su[(size_t)u * D + c + 1], b.y);
  if (lane == 0) {
    atomicAdd(&cm[m], 1.0f);
    atomicAdd(&cu[u], 1.0f);
  }
}

// ---------------------------------------------------------------------------
// Kernel 3: fused SAGE transform with fp32 WMMA.
//   out[r,:] = act( (s[r,:]/max(cnt[r],1)) @ Wl + bias + xdst[r,:] @ Wr )
// Block: 256 threads = 8 waves -> 32 rows x 64 cols.
// Each wave: one 16x16 tile, 32x V_WMMA_F32_16X16X4_F32 (two fused GEMMs).
// ---------------------------------------------------------------------------
__global__ __launch_bounds__(256) void sage_wmma_kernel(
    const float* __restrict__ s, const float* __restrict__ cnt,
    const float* __restrict__ xdst,
    const float* __restrict__ Wl, const float* __restrict__ Wr,
    const float* __restrict__ bias,
    float* __restrict__ out, int nrows, int do_relu) {
  __shared__ float ldsWl[D * WSTRIDE];
  __shared__ float ldsWr[D * WSTRIDE];
  __shared__ float ldsMean[32 * ASTRIDE];
  __shared__ float ldsX[32 * ASTRIDE];

  const int tid  = threadIdx.x;
  const int row0 = blockIdx.x * 32;

  // Stage weights: 4096 floats each, 4 float4 per thread per matrix.
  for (int i = tid * 4; i < D * D; i += 256 * 4) {
    const int r = i >> 6, c = i & 63;
    *(float4*)&ldsWl[r * WSTRIDE + c] = *(const float4*)&Wl[i];
    *(float4*)&ldsWr[r * WSTRIDE + c] = *(const float4*)&Wr[i];
  }
  // Stage 32 rows of mean (s/cnt) and x_dst.
  for (int i = tid * 4; i < 32 * D; i += 256 * 4) {
    const int r = i >> 6, c = i & 63;
    const int g = row0 + r;
    float4 mv = make_float4(0.f, 0.f, 0.f, 0.f);
    float4 xv = make_float4(0.f, 0.f, 0.f, 0.f);
    if (g < nrows) {
      const float inv = 1.0f / fmaxf(cnt[g], 1.0f);
      const float4 sv = *(const float4*)&s[(size_t)g * D + c];
      mv = make_float4(sv.x * inv, sv.y * inv, sv.z * inv, sv.w * inv);
      xv = *(const float4*)&xdst[(size_t)g * D + c];
    }
    *(float4*)&ldsMean[r * ASTRIDE + c] = mv;
    *(float4*)&ldsX[r * ASTRIDE + c]    = xv;
  }
  __syncthreads();

  const int wave  = tid >> 5;
  const int lane  = tid & 31;
  const int rt    = wave >> 2;          // row tile 0..1
  const int ct    = wave & 3;           // col tile 0..3
  const int mrow  = rt * 16 + (lane & 15);   // A-matrix row for this lane
  const int ncol  = ct * 16 + (lane & 15);   // B/C/D column for this lane
  const int khalf = (lane >> 4) * 2;         // lanes 16-31 hold K+2,K+3

  v8f acc = {0.f, 0.f, 0.f, 0.f, 0.f, 0.f, 0.f, 0.f};
  // GEMM 1: mean @ Wl
  #pragma unroll
  for (int k = 0; k < D; k += 4) {
    const int kk = k + khalf;
    const v2f a = *(const v2f*)&ldsMean[mrow * ASTRIDE + kk];
    v2f b;
    b.x = ldsWl[(kk + 0) * WSTRIDE + ncol];
    b.y = ldsWl[(kk + 1) * WSTRIDE + ncol];
    acc = __builtin_amdgcn_wmma_f32_16x16x4_f32(false, a, false, b,
                                                (short)0, acc, false, false);
  }
  // GEMM 2: x_dst @ Wr (same accumulator)
  #pragma unroll
  for (int k = 0; k < D; k += 4) {
    const int kk = k + khalf;
    const v2f a = *(const v2f*)&ldsX[mrow * ASTRIDE + kk];
    v2f b;
    b.x = ldsWr[(kk + 0) * WSTRIDE + ncol];
    b.y = ldsWr[(kk + 1) * WSTRIDE + ncol];
    acc = __builtin_amdgcn_wmma_f32_16x16x4_f32(false, a, false, b,
                                                (short)0, acc, false, false);
  }

  // Epilogue: bias (+ReLU), write D-matrix per ISA layout:
  // VGPR v holds M = v (lanes 0-15) / M = v+8 (lanes 16-31), N = lane&15.
  const float bcol  = bias[ncol];
  const int   mhalf = (lane < 16) ? 0 : 8;
  #pragma unroll
  for (int v = 0; v < 8; ++v) {
    const int g = row0 + rt * 16 + mhalf + v;
    if (g < nrows) {
      float val = acc[v] + bcol;
      if (do_relu) val = fmaxf(val, 0.0f);
      out[(size_t)g * D + ncol] = val;
    }
  }
}

// ---------------------------------------------------------------------------
// Launcher: full 2-layer pipeline, graph-capture safe (all work on `stream`).
// ---------------------------------------------------------------------------
extern "C" void kernel_launch(void* const* d_in, const int* in_sizes, int n_in,
                              void* d_out, int out_size, void* d_ws, size_t ws_size,
                              hipStream_t stream) {
  const float* x_user  = (const float*)d_in[0];
  const float* x_movie = (const float*)d_in[1];
  const float* Wl1_um  = (const float*)d_in[2];
  const float* Wr1_um  = (const float*)d_in[3];
  const float* b1_um   = (const float*)d_in[4];
  const float* Wl1_mu  = (const float*)d_in[5];
  const float* Wr1_mu  = (const float*)d_in[6];
  const float* b1_mu   = (const float*)d_in[7];
  const float* Wl2_um  = (const float*)d_in[8];
  const float* Wr2_um  = (const float*)d_in[9];
  const float* b2_um   = (const float*)d_in[10];
  const float* Wl2_mu  = (const float*)d_in[11];
  const float* Wr2_mu  = (const float*)d_in[12];
  const float* b2_mu   = (const float*)d_in[13];
  const int*   uidx    = (const int*)d_in[14];
  const int*   midx    = (const int*)d_in[15];

  float* ws = (float*)d_ws;
  float* su = ws + OFF_SU;
  float* sm = ws + OFF_SM;
  float* cu = ws + OFF_CU;
  float* cm = ws + OFF_CM;
  float* u1 = ws + OFF_U1;
  float* m1 = ws + OFF_M1;

  float* u2 = (float*)d_out;                       // first N_USER rows
  float* m2 = (float*)d_out + (size_t)N_USER * D;  // then N_MOVIE rows

  const dim3 blk(256);
  const dim3 gZero((ZERO_F4 + 255) / 256);
  const dim3 gScat((unsigned)(((size_t)N_EDGE * 32 + 255) / 256));
  const dim3 gMov(N_MOVIE / 32);   // 100000 % 32 == 0 -> 3125 blocks
  const dim3 gUsr(N_USER / 32);    // 200000 % 32 == 0 -> 6250 blocks

  // ---- layer 1 ----
  zero_f4_kernel<<<gZero, blk, 0, stream>>>((float4*)ws, ZERO_F4);
  scatter_kernel<<<gScat, blk, 0, stream>>>(x_user, x_movie, uidx, midx,
                                            su, sm, cu, cm, N_EDGE);
  sage_wmma_kernel<<<gMov, blk, 0, stream>>>(sm, cm, x_movie, Wl1_um, Wr1_um,
                                             b1_um, m1, N_MOVIE, 1);
  sage_wmma_kernel<<<gUsr, blk, 0, stream>>>(su, cu, x_user, Wl1_mu, Wr1_mu,
                                             b1_mu, u1, N_USER, 1);

  // ---- layer 2 ----
  zero_f4_kernel<<<gZero, blk, 0, stream>>>((float4*)ws, ZERO_F4);
  scatter_kernel<<<gScat, blk, 0, stream>>>(u1, m1, uidx, midx,
                                            su, sm, cu, cm, N_EDGE);
  sage_wmma_kernel<<<gMov, blk, 0, stream>>>(sm, cm, m1, Wl2_um, Wr2_um,
                                             b2_um, m2, N_MOVIE, 0);
  sage_wmma_kernel<<<gUsr, blk, 0, stream>>>(su, cu, u1, Wl2_mu, Wr2_mu,
                                             b2_mu, u2, N_USER, 0);
}